// DistanceLoss_14310831030656
// MI455X (gfx1250) — compile-verified
//
#include <hip/hip_runtime.h>

// ---------------------------------------------------------------------------
// PairwiseDistance(p=2, eps=1e-6) sum-reduction for [16,8,512,512] f32.
//   loss = sum_{b,c,h} sqrt( sum_w (o[b,c,h,w] - t[b,c,h,w] + eps)^2 )
//
// Memory-bound: 268 MB read @ 23.3 TB/s => ~11.5 us floor. Pass 1 streams
// with non-temporal b128 loads (read-once data; don't pollute the 192MB L2).
// Pass 2 reduces 4096 block partials with a 256-thread block using b128
// loads (pipelined, not serialized), and finishes the last 32-lane reduce
// on the CDNA5 matrix pipe via v_wmma_f32_16x16x4_f32 with an all-ones B.
// ---------------------------------------------------------------------------

typedef float f32x4 __attribute__((ext_vector_type(4)));
typedef float v2f   __attribute__((ext_vector_type(2)));
typedef float v8f   __attribute__((ext_vector_type(8)));

#define WAVE            32
#define WAVES_PER_BLOCK 8
#define BLOCK           (WAVE * WAVES_PER_BLOCK)
#define ROW_W           512          // last-dim length (reference is fixed-shape)
#define EPS_F           1e-6f
#define P2_BLOCK        256

// ---------------------------------------------------------------------------
// Pass 1: one wave per row (grid-stride over rows). Each lane streams 4
// float4 chunks per input row (coalesced: lane l reads bytes [16l,16l+16)
// of each 512B row-chunk), accumulates sum of squares, wave-reduces, takes
// sqrt, accumulates row norms; block writes a single partial to d_ws.
// ---------------------------------------------------------------------------
__global__ __launch_bounds__(BLOCK) void pdist_pass1(
    const float* __restrict__ out_p,
    const float* __restrict__ tgt_p,
    float* __restrict__ partial,
    int rows)
{
    __shared__ float lds[WAVES_PER_BLOCK];

    const int lane   = threadIdx.x & (WAVE - 1);
    const int wave   = threadIdx.x >> 5;
    const int gwave  = blockIdx.x * WAVES_PER_BLOCK + wave;
    const int nwaves = gridDim.x * WAVES_PER_BLOCK;

    float acc = 0.0f;   // sum of row norms handled by this wave (replicated per lane)

    for (int row = gwave; row < rows; row += nwaves) {
        const f32x4* o4 = (const f32x4*)(out_p + (size_t)row * ROW_W);
        const f32x4* t4 = (const f32x4*)(tgt_p + (size_t)row * ROW_W);

        float s = 0.0f;
        #pragma unroll
        for (int c = 0; c < ROW_W / (WAVE * 4); ++c) {   // 4 chunks of 128 floats
            const int idx = lane + c * WAVE;
            f32x4 ov = __builtin_nontemporal_load(&o4[idx]);  // read-once stream: NT hint
            f32x4 tv = __builtin_nontemporal_load(&t4[idx]);
            f32x4 d  = ov - tv + EPS_F;
            s += d.x * d.x + d.y * d.y + d.z * d.z + d.w * d.w;
        }

        // wave32 all-reduce of the row's sum of squares
        #pragma unroll
        for (int off = 16; off >= 1; off >>= 1)
            s += __shfl_xor(s, off, WAVE);

        acc += sqrtf(s);
    }

    if (lane == 0) lds[wave] = acc;
    __syncthreads();

    if (threadIdx.x == 0) {
        float b = 0.0f;
        #pragma unroll
        for (int i = 0; i < WAVES_PER_BLOCK; ++i) b += lds[i];
        partial[blockIdx.x] = b;
    }
}

// ---------------------------------------------------------------------------
// Pass 2: single 256-thread block. Vectorized b128 reads of the partials
// (4 independent loads/thread for n=4096 -> latency hidden by MLP, unlike a
// serialized one-wave scalar loop). Per-wave shuffle reduce -> 8 wave sums
// in LDS -> wave 0 (uniform branch: EXEC all-ones, as WMMA requires)
// finishes with the WMMA ones-matrix reduction:
//   A (16x4 f32): a0 = lane value, a1 = 0 (each lane value occupies exactly
//                 one A slot; zeros elsewhere -> layout-detail independent)
//   B (4x16 f32): all ones (layout-invariant)
//   D[m,n] = sum_k A[m,k]; per documented C/D layout, summing the 8 D regs
//   per lane and adding the opposite lane-group half gives the full sum.
// ---------------------------------------------------------------------------
__global__ __launch_bounds__(P2_BLOCK) void pdist_pass2(
    const float* __restrict__ partial,
    int n,
    float* __restrict__ out)
{
    __shared__ float lds[P2_BLOCK / WAVE];

    const int tid  = threadIdx.x;
    const int lane = tid & (WAVE - 1);
    const int wave = tid >> 5;

    float s = 0.0f;

    const int n4 = n >> 2;                       // n = 4096 -> 1024 vec4s
    const f32x4* p4 = (const f32x4*)partial;
    #pragma unroll 4
    for (int i = tid; i < n4; i += P2_BLOCK) {   // 4 indep b128 loads/thread
        f32x4 v = p4[i];
        s += v.x + v.y + v.z + v.w;
    }
    for (int i = (n4 << 2) + tid; i < n; i += P2_BLOCK)  // safety tail
        s += partial[i];

    #pragma unroll
    for (int off = 16; off >= 1; off >>= 1)
        s += __shfl_xor(s, off, WAVE);

    if (lane == 0) lds[wave] = s;
    __syncthreads();

    if (wave == 0) {                             // uniform per-wave: EXEC all-ones
        float v = (lane < P2_BLOCK / WAVE) ? lds[lane] : 0.0f;

#if __has_builtin(__builtin_amdgcn_wmma_f32_16x16x4_f32)
        v2f a; a.x = v;    a.y = 0.0f;
        v2f b; b.x = 1.0f; b.y = 1.0f;
        v8f c = {};
        // 8 args: (neg_a, A, neg_b, B, c_mod, C, reuse_a, reuse_b)
        v8f d = __builtin_amdgcn_wmma_f32_16x16x4_f32(
            false, a, false, b, (short)0, c, false, false);

        float half = d[0] + d[1] + d[2] + d[3] + d[4] + d[5] + d[6] + d[7];
        float tot  = half + __shfl_xor(half, 16, WAVE);
#else
        float tot = v;
        #pragma unroll
        for (int off = 16; off >= 1; off >>= 1)
            tot += __shfl_xor(tot, off, WAVE);
#endif

        if (lane == 0) out[0] = tot;
    }
}

// ---------------------------------------------------------------------------
// Launch: deterministic two-pass (no float atomics -> bit-stable replays).
// ---------------------------------------------------------------------------
extern "C" void kernel_launch(void* const* d_in, const int* in_sizes, int n_in,
                              void* d_out, int out_size, void* d_ws, size_t ws_size,
                              hipStream_t stream) {
    const float* o = (const float*)d_in[0];
    const float* t = (const float*)d_in[1];
    float* out     = (float*)d_out;
    float* partial = (float*)d_ws;

    const int n    = in_sizes[0];       // 16*8*512*512 = 33,554,432
    const int rows = n / ROW_W;         // 65,536 rows of 512 floats

    int blocks = 4096;                  // 32,768 waves -> exactly 2 rows/wave
    const int max_blocks = (rows + WAVES_PER_BLOCK - 1) / WAVES_PER_BLOCK;
    if (blocks > max_blocks) blocks = max_blocks;

    pdist_pass1<<<blocks, BLOCK, 0, stream>>>(o, t, partial, rows);
    pdist_pass2<<<1, P2_BLOCK, 0, stream>>>(partial, blocks, out);
}